// CenterLoss_7103875908251
// MI455X (gfx1250) — compile-verified
//
#include <hip/hip_runtime.h>
#include <hip/hip_bf16.h>

typedef __attribute__((ext_vector_type(2))) float v2f;
typedef __attribute__((ext_vector_type(8))) float v8f;

#define DIMS     512
#define NSAMP    16384
#define NBLOCKS  512     // 2 waves/block * 16 samples/wave * 512 = 16384

// One wave computes 16 samples' squared distances using V_WMMA_F32_16X16X4_F32
// with A == B (same registers), so diag(D) = per-row sum of squares of delta.
__global__ __launch_bounds__(64) void center_loss_wmma_kernel(
    const float* __restrict__ features,
    const float* __restrict__ centers,
    const int*   __restrict__ labels,
    float*       __restrict__ partials)
{
    const int lane = threadIdx.x & 31;
    const int wave = threadIdx.x >> 5;

    // 16-sample tile handled by this wave
    const int tileBase = (blockIdx.x * 2 + wave) * 16;
    const int m        = lane & 15;     // A-matrix row M for this lane (both halves)
    const int half     = lane >> 4;     // 0 => low K pair slot, 1 => high K pair slot
    const int sample   = tileBase + m;

    const int lbl = labels[sample];
    const float* __restrict__ frow = features + (size_t)sample * DIMS;
    const float* __restrict__ crow = centers  + (size_t)lbl    * DIMS;

    // K-order inside a dot product is free: give each lane an aligned float4.
    // Per 8-wide K group: lanes 0-15 own k..k+3, lanes 16-31 own k+4..k+7.
    const int colBase = half * 4;

    v8f acc = {};

    #pragma unroll 4
    for (int k = 0; k < DIMS; k += 8) {
        const float4 f = *(const float4*)(frow + k + colBase);
        const float4 c = *(const float4*)(crow + k + colBase);
        v2f a0, a1;
        a0.x = f.x - c.x;  a0.y = f.y - c.y;
        a1.x = f.z - c.z;  a1.y = f.w - c.w;
        // D = A*A^T accumulate: diagonal collects sum of squared deltas.
        acc = __builtin_amdgcn_wmma_f32_16x16x4_f32(false, a0, false, a0,
                                                    (short)0, acc, false, false);
        acc = __builtin_amdgcn_wmma_f32_16x16x4_f32(false, a1, false, a1,
                                                    (short)0, acc, false, false);
    }

    // Diagonal of 16x16 f32 C/D layout:
    //   lanes 0-7  hold D[l][l]       in acc[l]
    //   lanes 24-31 hold D[l-16][l-16] in acc[l-24]
    const bool has = (lane < 8) || (lane >= 24);
    const int  p   = (lane < 8) ? lane : (lane - 24);  // negative for non-diag lanes
    float v = 0.0f;
    #pragma unroll
    for (int j = 0; j < 8; ++j) v = (p == j) ? acc[j] : v;

    v = fminf(fmaxf(v, 1e-12f), 1e12f);   // reference's clip, applied per sample
    v = has ? v : 0.0f;

    // wave32 tree sum of the 16 diagonal values
    #pragma unroll
    for (int off = 16; off >= 1; off >>= 1)
        v += __shfl_xor(v, off, 32);

    __shared__ float wsum[2];
    if (lane == 0) wsum[wave] = v;
    __syncthreads();
    if (threadIdx.x == 0) partials[blockIdx.x] = wsum[0] + wsum[1];
}

// Deterministic fixed-order reduction of the 512 block partials.
__global__ __launch_bounds__(256) void center_loss_reduce_kernel(
    const float* __restrict__ partials,
    float*       __restrict__ out)
{
    __shared__ float s[256];
    const int t = threadIdx.x;
    s[t] = partials[t] + partials[t + 256];
    __syncthreads();
    #pragma unroll
    for (int off = 128; off > 0; off >>= 1) {
        if (t < off) s[t] += s[t + off];
        __syncthreads();
    }
    if (t == 0) out[0] = s[0] * (1.0f / (float)NSAMP);
}

extern "C" void kernel_launch(void* const* d_in, const int* in_sizes, int n_in,
                              void* d_out, int out_size, void* d_ws, size_t ws_size,
                              hipStream_t stream) {
    (void)in_sizes; (void)n_in; (void)out_size; (void)ws_size;
    const float* features = (const float*)d_in[0];
    const float* centers  = (const float*)d_in[1];
    const int*   labels   = (const int*)d_in[2];
    float* partials = (float*)d_ws;     // 512 floats of scratch
    float* out      = (float*)d_out;

    center_loss_wmma_kernel<<<NBLOCKS, 64, 0, stream>>>(features, centers, labels, partials);
    center_loss_reduce_kernel<<<1, 256, 0, stream>>>(partials, out);
}